// GCNEncoder_77979426226625
// MI455X (gfx1250) — compile-verified
//
#include <hip/hip_runtime.h>

typedef __attribute__((ext_vector_type(2))) float v2f;
typedef __attribute__((ext_vector_type(8))) float v8f;

#define LDS_STRIDE 130   // 128 + 2 pad: avoids 16-way bank conflict on 64 banks

// ---------------- degree accumulation (once, reused by all 3 layers) --------
__global__ void gcn_degree_kernel(const int* __restrict__ src,
                                  const int* __restrict__ dst,
                                  float* __restrict__ deg_out,
                                  float* __restrict__ deg_in, int E) {
    int e = blockIdx.x * blockDim.x + threadIdx.x;
    if (e < E) {
        unsafeAtomicAdd(&deg_out[src[e]], 1.0f);
        unsafeAtomicAdd(&deg_in[dst[e]], 1.0f);
    }
}

// in-place deg -> rsqrt(max(deg,1))
__global__ void gcn_norm_kernel(float* __restrict__ d, int n) {
    int i = blockIdx.x * blockDim.x + threadIdx.x;
    if (i < n) d[i] = rsqrtf(fmaxf(d[i], 1.0f));
}

// ---------------- SpMM scatter: agg[dst] += h[src] * norm_src[src] ----------
// 1 edge per wave: 32 lanes x float4 = 128 floats; gather is one coalesced
// 512B row read (L2-resident: whole feature matrix is 51MB < 192MB L2).
__global__ __launch_bounds__(256) void gcn_spmm_kernel(
        const float* __restrict__ h, const int* __restrict__ src,
        const int* __restrict__ dst, const float* __restrict__ norm_src,
        float* __restrict__ agg, int E) {
    int lane  = threadIdx.x & 31;
    int eslot = threadIdx.x >> 5;
    int e = blockIdx.x * 8 + eslot;
    if (e >= E) return;
    int s = src[e];
    int d = dst[e];
    float ns = norm_src[s];
    float4 v = ((const float4*)(h + (size_t)s * 128))[lane];
    float* ap = agg + (size_t)d * 128 + lane * 4;
    unsafeAtomicAdd(ap + 0, v.x * ns);
    unsafeAtomicAdd(ap + 1, v.y * ns);
    unsafeAtomicAdd(ap + 2, v.z * ns);
    unsafeAtomicAdd(ap + 3, v.w * ns);
}

// ---------------- GEMM: out = (agg * norm_dst[:,None]) @ W + b --------------
// One block = 16 rows of output. A tile (16x128, scaled by norm_dst) staged in
// LDS. Each wave owns a 16-column tile; K=128 -> 32 x v_wmma_f32_16x16x4_f32.
template <int NOUT>
__global__ __launch_bounds__(32 * (NOUT / 16)) void gcn_gemm_wmma_kernel(
        const float* __restrict__ A,      // [N,128] aggregated features
        const float* __restrict__ normd,  // [N] dest-degree norm
        const float* __restrict__ W,      // [128, NOUT]
        const float* __restrict__ bias,   // [NOUT]
        float* __restrict__ out,          // [N, NOUT]
        int N) {
    __shared__ float lds_a[16 * LDS_STRIDE];

    const int rowbase = blockIdx.x * 16;

    // Cooperative load of the 16x128 A tile, fused with norm_dst scaling.
    for (int idx = threadIdx.x; idx < 16 * 32; idx += blockDim.x) {
        int r  = idx >> 5;   // tile row 0..15
        int c4 = idx & 31;   // float4 column
        int row = rowbase + r;
        float4 v = make_float4(0.f, 0.f, 0.f, 0.f);
        float nd = 0.f;
        if (row < N) {
            v  = ((const float4*)(A + (size_t)row * 128))[c4];
            nd = normd[row];
        }
        float* p = &lds_a[r * LDS_STRIDE + c4 * 4];
        p[0] = v.x * nd; p[1] = v.y * nd; p[2] = v.z * nd; p[3] = v.w * nd;
    }
    __syncthreads();

    const int wave    = threadIdx.x >> 5;       // wave32
    const int lane    = threadIdx.x & 31;
    const int colbase = wave * 16;
    const int n       = colbase + (lane & 15);  // output column for B/C/D
    const int m       = lane & 15;              // A row for this lane
    const int koff    = (lane >> 4) * 2;        // lanes 16-31 carry K+2 (ISA 7.12.2)

    v8f c = {};
    for (int k0 = 0; k0 < 128; k0 += 4) {
        v2f a, b;
        a.x = lds_a[m * LDS_STRIDE + k0 + koff];       // A[m][k], A[m][k+1]
        a.y = lds_a[m * LDS_STRIDE + k0 + koff + 1];
        b.x = W[(k0 + koff)     * NOUT + n];           // B[k][n], B[k+1][n]
        b.y = W[(k0 + koff + 1) * NOUT + n];
        // D = A x B + C  (16x16x4 fp32 WMMA)
        c = __builtin_amdgcn_wmma_f32_16x16x4_f32(
                /*neg_a=*/false, a, /*neg_b=*/false, b,
                /*c_mod=*/(short)0, c, /*reuse_a=*/false, /*reuse_b=*/false);
    }

    // C/D layout: VGPR v -> row rowbase+v (lanes 0-15) or rowbase+v+8 (16-31)
    float bv = bias[n];
    int radd = (lane >> 4) * 8;
#pragma unroll
    for (int v = 0; v < 8; ++v) {
        int row = rowbase + v + radd;
        if (row < N) out[(size_t)row * NOUT + n] = c[v] + bv;
    }
}

// ---------------------------------------------------------------------------
extern "C" void kernel_launch(void* const* d_in, const int* in_sizes, int n_in,
                              void* d_out, int out_size, void* d_ws, size_t ws_size,
                              hipStream_t stream) {
    const float* feat = (const float*)d_in[0];
    const float* W0   = (const float*)d_in[1];
    const float* b0   = (const float*)d_in[2];
    const float* W1   = (const float*)d_in[3];
    const float* b1   = (const float*)d_in[4];
    const float* W2   = (const float*)d_in[5];
    const float* b2   = (const float*)d_in[6];
    const int*   src  = (const int*)d_in[7];
    const int*   dst  = (const int*)d_in[8];

    const int N = in_sizes[0] / 128;
    const int E = in_sizes[7];

    float* ws   = (float*)d_ws;
    float* ns   = ws;                       // [N] src-degree norm
    float* nd   = ws + N;                   // [N] dst-degree norm
    float* bufA = ws + 2 * (size_t)N;       // [N,128] aggregation accumulator
    float* bufB = bufA + (size_t)N * 128;   // [N,128] layer output
    float* out  = (float*)d_out;            // [N,64]

    const size_t featBytes = (size_t)N * 128 * sizeof(float);
    const int eBlocks    = (E + 255) / 256;
    const int spmmBlocks = (E + 7) / 8;
    const int rowBlocks  = (N + 15) / 16;

    // degrees -> norms (computed once, shared by all layers)
    hipMemsetAsync(ns, 0, 2 * (size_t)N * sizeof(float), stream);
    gcn_degree_kernel<<<eBlocks, 256, 0, stream>>>(src, dst, ns, nd, E);
    gcn_norm_kernel<<<(2 * N + 255) / 256, 256, 0, stream>>>(ws, 2 * N);

    // layer 1: feat -> bufB
    hipMemsetAsync(bufA, 0, featBytes, stream);
    gcn_spmm_kernel<<<spmmBlocks, 256, 0, stream>>>(feat, src, dst, ns, bufA, E);
    gcn_gemm_wmma_kernel<128><<<rowBlocks, 256, 0, stream>>>(bufA, nd, W0, b0, bufB, N);

    // layer 2: bufB -> bufB
    hipMemsetAsync(bufA, 0, featBytes, stream);
    gcn_spmm_kernel<<<spmmBlocks, 256, 0, stream>>>(bufB, src, dst, ns, bufA, E);
    gcn_gemm_wmma_kernel<128><<<rowBlocks, 256, 0, stream>>>(bufA, nd, W1, b1, bufB, N);

    // layer 3: bufB -> d_out (128 -> 64)
    hipMemsetAsync(bufA, 0, featBytes, stream);
    gcn_spmm_kernel<<<spmmBlocks, 256, 0, stream>>>(bufB, src, dst, ns, bufA, E);
    gcn_gemm_wmma_kernel<64><<<rowBlocks, 128, 0, stream>>>(bufA, nd, W2, b2, out, N);
}